// MultiTypeGNN_43233140801908
// MI455X (gfx1250) — compile-verified
//
#include <hip/hip_runtime.h>
#include <math.h>

#define NPT    25000
#define TYPES  4
#define NTOT   (NPT*TYPES)      // 100000
#define EDGES  600000
#define HDIM   128
#define RREL   8
#define NHEAD  4
#define DKH    32               // HDIM / NHEAD
#define NLAYER 2
#define KC     32               // GEMM K-chunk staged in LDS

typedef float v2f __attribute__((ext_vector_type(2)));
typedef float v8f __attribute__((ext_vector_type(8)));

// ---------------------------------------------------------------------------
// Async-copy one KC x 64 panel of W into LDS (CDNA5 ASYNCcnt path).
// 2048 floats = 512 x b128; 256 threads issue 2 lane-loads each.
// ---------------------------------------------------------------------------
__device__ __forceinline__ void stage_panel(const float* __restrict__ W, int N,
                                            int n0, int kc, unsigned ldsBase, int tid)
{
    #pragma unroll
    for (int p = 0; p < 2; ++p) {
        const int idx = tid + p * 256;
        const int r   = idx >> 4;          // 0..31 (K row within chunk)
        const int c4  = (idx & 15) << 2;   // 0..60 (col, float4 granularity)
        unsigned long long ga =
            (unsigned long long)(uintptr_t)(W + (size_t)(kc + r) * N + n0 + c4);
        unsigned la = ldsBase + (unsigned)idx * 16u;
        asm volatile("global_load_async_to_lds_b128 %0, %1, off"
                     :: "v"(la), "v"(ga) : "memory");
    }
}

// ---------------------------------------------------------------------------
// GEMM: C[M,N] = act(A[M,K] @ W[K,N] + bias[N]);  act: 0=none, 1=relu
// Block = 256 thr = 8 waves; block tile 128(M) x 64(N).
// Wave w: rows m0 = blockRow*128 + w*16, 4 WMMA accumulators over 64 cols.
// W panel double-buffered in LDS via global_load_async_to_lds_b128.
// A fragments for a whole K-chunk prefetched to registers (8 x float2).
// Requires: K % 32 == 0, N % 64 == 0 (true for all calls here).
// ---------------------------------------------------------------------------
__global__ __launch_bounds__(256)
void gemm_bias_act(const float* __restrict__ A, const float* __restrict__ W,
                   const float* __restrict__ bias, float* __restrict__ C,
                   int M, int K, int N, int act)
{
    __shared__ __align__(16) float sW[2][KC * 64];   // 16 KB

    const int tid  = threadIdx.x;
    const int wave = tid >> 5;
    const int lane = tid & 31;
    const int half = lane >> 4;       // 0: lanes 0-15, 1: lanes 16-31
    const int l16  = lane & 15;

    const int panels  = N >> 6;
    const int panel   = blockIdx.x % panels;
    const int rowBlk  = blockIdx.x / panels;
    const int n0      = panel << 6;
    const int m0      = rowBlk * 128 + wave * 16;

    const int  rowA      = m0 + l16;
    const bool rowAvalid = (rowA < M);
    const float* Arow    = A + (size_t)(rowAvalid ? rowA : (M - 1)) * K;

    const unsigned ldsB0 = (unsigned)(uintptr_t)&sW[0][0];
    const unsigned ldsB1 = (unsigned)(uintptr_t)&sW[1][0];

    v8f acc[4];
    #pragma unroll
    for (int c = 0; c < 4; ++c) acc[c] = (v8f){0.f,0.f,0.f,0.f,0.f,0.f,0.f,0.f};

    const int nChunks = K / KC;
    stage_panel(W, N, n0, 0, ldsB0, tid);

    for (int ch = 0; ch < nChunks; ++ch) {
        const int kc = ch * KC;

        // Prefetch this chunk's A fragments (one clause of 8 x b64).
        v2f aFrag[KC / 4];
        #pragma unroll
        for (int ks = 0; ks < KC; ks += 4) {
            float2 av = *(const float2*)(Arow + kc + ks + 2 * half);
            aFrag[ks >> 2].x = rowAvalid ? av.x : 0.f;
            aFrag[ks >> 2].y = rowAvalid ? av.y : 0.f;
        }

        if (ch + 1 < nChunks) {
            stage_panel(W, N, n0, kc + KC, ((ch + 1) & 1) ? ldsB1 : ldsB0, tid);
            asm volatile("s_wait_asynccnt 0x2" ::: "memory");  // this chunk done
        } else {
            asm volatile("s_wait_asynccnt 0x0" ::: "memory");
        }
        __syncthreads();

        const float* Wb = &sW[ch & 1][0];
        #pragma unroll
        for (int ks = 0; ks < KC; ks += 4) {
            const v2f a = aFrag[ks >> 2];
            const int rb = (ks + 2 * half) * 64 + l16;
            #pragma unroll
            for (int c = 0; c < 4; ++c) {
                v2f b;
                b.x = Wb[rb + c * 16];        // K row ks+2h
                b.y = Wb[rb + 64 + c * 16];   // K row ks+2h+1
                acc[c] = __builtin_amdgcn_wmma_f32_16x16x4_f32(
                             false, a, false, b, (short)0, acc[c], false, false);
            }
        }
        __syncthreads();   // safe to overwrite this buffer next round
    }

    #pragma unroll
    for (int c = 0; c < 4; ++c) {
        const int col = n0 + c * 16 + l16;
        const float bv = bias ? bias[col] : 0.f;
        #pragma unroll
        for (int r = 0; r < 8; ++r) {
            const int row = m0 + r + 8 * half;   // C layout: VGPR r -> M=r (+8 hi lanes)
            if (row < M) {
                float v = acc[c][r] + bv;
                if (act == 1) v = fmaxf(v, 0.f);
                C[(size_t)row * N + col] = v;
            }
        }
    }
}

// ---------------------------------------------------------------------------
__global__ void fill_f32(float* __restrict__ p, float v, long n)
{
    long i = (long)blockIdx.x * blockDim.x + threadIdx.x;
    if (i < n) p[i] = v;
}

__global__ void gelu_inplace(float* __restrict__ p, long n)
{
    long i = (long)blockIdx.x * blockDim.x + threadIdx.x;
    if (i < n) {
        float x = p[i];
        p[i] = 0.5f * x * (1.f + erff(x * 0.70710678118654752f));
    }
}

// ---------------------------------------------------------------------------
// Pass A: att[e,h] = (q_d . (k_s @ ra[r,h])) * rp[r,h] / sqrt(DK); amax = segmax
// grid.y = head; relation matrices for this head staged into LDS (32 KB).
// ---------------------------------------------------------------------------
__global__ __launch_bounds__(256)
void edge_att(const float* __restrict__ q, const float* __restrict__ k,
              const int* __restrict__ src, const int* __restrict__ dst,
              const int* __restrict__ et,
              const float* __restrict__ ra,   // [R,NHEAD,DK,DK] (layer slice)
              const float* __restrict__ rp,   // [R,NHEAD]
              float* __restrict__ att,        // [E,NHEAD]
              float* __restrict__ amax,       // [N,NHEAD]
              int E)
{
    __shared__ float sRA[RREL * DKH * DKH];   // 8192 floats = 32 KB
    const int head = blockIdx.y;
    for (int i = threadIdx.x; i < RREL * DKH * DKH; i += blockDim.x) {
        int r  = i >> 10;
        int df = i & 1023;
        sRA[i] = ra[((long)r * NHEAD + head) * (DKH * DKH) + df];
    }
    __syncthreads();

    const int e = blockIdx.x * blockDim.x + threadIdx.x;
    if (e >= E) return;
    const int s  = src[e];
    const int d_ = dst[e];
    const int r  = et[e];

    const float4* kv4 = (const float4*)(k + (long)s  * HDIM + head * DKH);
    const float4* qv4 = (const float4*)(q + (long)d_ * HDIM + head * DKH);
    float kreg[DKH], qreg[DKH];
    #pragma unroll
    for (int i = 0; i < DKH/4; ++i) {
        float4 t = kv4[i];
        kreg[4*i+0]=t.x; kreg[4*i+1]=t.y; kreg[4*i+2]=t.z; kreg[4*i+3]=t.w;
        float4 u = qv4[i];
        qreg[4*i+0]=u.x; qreg[4*i+1]=u.y; qreg[4*i+2]=u.z; qreg[4*i+3]=u.w;
    }

    const float* M = sRA + r * DKH * DKH;
    float a = 0.f;
    #pragma unroll 4
    for (int f = 0; f < DKH; ++f) {
        float kt = 0.f;
        #pragma unroll
        for (int d0 = 0; d0 < DKH; ++d0) kt += kreg[d0] * M[d0 * DKH + f];
        a += qreg[f] * kt;
    }
    a *= rp[r * NHEAD + head] * 0.17677669529663687f;   // 1/sqrt(32)

    att[(long)e * NHEAD + head] = a;
    atomicMax(&amax[(long)d_ * NHEAD + head], a);
}

// ---------------------------------------------------------------------------
// Pass B: ea = exp(att - amax[dst]); den[dst] += ea   (in-place on att)
// ---------------------------------------------------------------------------
__global__ void edge_exp(float* __restrict__ att, const int* __restrict__ dst,
                         const float* __restrict__ amax, float* __restrict__ den,
                         int EH)
{
    int i = blockIdx.x * blockDim.x + threadIdx.x;
    if (i >= EH) return;
    int e = i >> 2;
    int h = i & 3;
    int d_ = dst[e];
    float x = expf(att[i] - amax[(long)d_ * NHEAD + h]);
    att[i] = x;
    atomicAdd(&den[(long)d_ * NHEAD + h], x);
}

// ---------------------------------------------------------------------------
// Pass C: res[dst, h*DK + f] += (ea/max(den,1e-9)) * (v_s @ rm[r,h])[f]
// ---------------------------------------------------------------------------
__global__ __launch_bounds__(256)
void edge_scatter(const float* __restrict__ v,
                  const int* __restrict__ src, const int* __restrict__ dst,
                  const int* __restrict__ et,
                  const float* __restrict__ rm,   // [R,NHEAD,DK,DK] (layer slice)
                  const float* __restrict__ ea,   // [E,NHEAD]
                  const float* __restrict__ den,  // [N,NHEAD]
                  float* __restrict__ res,        // [N,HDIM]
                  int E)
{
    __shared__ float sRM[RREL * DKH * DKH];
    const int head = blockIdx.y;
    for (int i = threadIdx.x; i < RREL * DKH * DKH; i += blockDim.x) {
        int r  = i >> 10;
        int df = i & 1023;
        sRM[i] = rm[((long)r * NHEAD + head) * (DKH * DKH) + df];
    }
    __syncthreads();

    const int e = blockIdx.x * blockDim.x + threadIdx.x;
    if (e >= E) return;
    const int s  = src[e];
    const int d_ = dst[e];
    const int r  = et[e];

    const float w = ea[(long)e * NHEAD + head] /
                    fmaxf(den[(long)d_ * NHEAD + head], 1e-9f);

    const float4* vv4 = (const float4*)(v + (long)s * HDIM + head * DKH);
    float vreg[DKH];
    #pragma unroll
    for (int i = 0; i < DKH/4; ++i) {
        float4 t = vv4[i];
        vreg[4*i+0]=t.x; vreg[4*i+1]=t.y; vreg[4*i+2]=t.z; vreg[4*i+3]=t.w;
    }

    const float* M = sRM + r * DKH * DKH;
    float* out = res + (long)d_ * HDIM + head * DKH;
    #pragma unroll 4
    for (int f = 0; f < DKH; ++f) {
        float mt = 0.f;
        #pragma unroll
        for (int d0 = 0; d0 < DKH; ++d0) mt += vreg[d0] * M[d0 * DKH + f];
        atomicAdd(&out[f], w * mt);
    }
}

// ---------------------------------------------------------------------------
// Skip-gate + LayerNorm: one wave32 per node row (4 elems per lane).
// ---------------------------------------------------------------------------
__global__ __launch_bounds__(128)
void node_update(const float* __restrict__ trans, const float* __restrict__ hin,
                 const float* __restrict__ skip,   // [T] (layer slice)
                 const float* __restrict__ gamma, const float* __restrict__ beta,
                 float* __restrict__ hout, int Nn)
{
    const int wave = threadIdx.x >> 5;
    const int lane = threadIdx.x & 31;
    const long n = (long)blockIdx.x * 4 + wave;
    if (n >= Nn) return;

    const int t = (int)(n / NPT);
    const float alpha = 1.f / (1.f + expf(-skip[t]));

    float4 tv = *(const float4*)(trans + n * HDIM + lane * 4);
    float4 hv = *(const float4*)(hin   + n * HDIM + lane * 4);
    float x[4];
    x[0] = tv.x*alpha + hv.x*(1.f-alpha);
    x[1] = tv.y*alpha + hv.y*(1.f-alpha);
    x[2] = tv.z*alpha + hv.z*(1.f-alpha);
    x[3] = tv.w*alpha + hv.w*(1.f-alpha);

    float sum = x[0]+x[1]+x[2]+x[3];
    #pragma unroll
    for (int off = 16; off > 0; off >>= 1) sum += __shfl_xor(sum, off, 32);
    const float mean = sum * (1.f / HDIM);

    float vs = 0.f;
    #pragma unroll
    for (int j = 0; j < 4; ++j) { float d = x[j]-mean; vs += d*d; }
    #pragma unroll
    for (int off = 16; off > 0; off >>= 1) vs += __shfl_xor(vs, off, 32);
    const float inv = rsqrtf(vs * (1.f / HDIM) + 1e-5f);

    float4 o;
    const int c = lane * 4;
    o.x = (x[0]-mean)*inv*gamma[c+0] + beta[c+0];
    o.y = (x[1]-mean)*inv*gamma[c+1] + beta[c+1];
    o.z = (x[2]-mean)*inv*gamma[c+2] + beta[c+2];
    o.w = (x[3]-mean)*inv*gamma[c+3] + beta[c+3];
    *(float4*)(hout + n * HDIM + c) = o;
}

// ---------------------------------------------------------------------------
extern "C" void kernel_launch(void* const* d_in, const int* in_sizes, int n_in,
                              void* d_out, int out_size, void* d_ws, size_t ws_size,
                              hipStream_t stream)
{
    const int IN_DIMS[TYPES] = {512, 256, 128, 64};

    const float* nf[TYPES]  = {(const float*)d_in[0], (const float*)d_in[1],
                               (const float*)d_in[2], (const float*)d_in[3]};
    const float* ew1[TYPES] = {(const float*)d_in[4],  (const float*)d_in[8],
                               (const float*)d_in[12], (const float*)d_in[16]};
    const float* eb1[TYPES] = {(const float*)d_in[5],  (const float*)d_in[9],
                               (const float*)d_in[13], (const float*)d_in[17]};
    const float* ew2[TYPES] = {(const float*)d_in[6],  (const float*)d_in[10],
                               (const float*)d_in[14], (const float*)d_in[18]};
    const float* eb2[TYPES] = {(const float*)d_in[7],  (const float*)d_in[11],
                               (const float*)d_in[15], (const float*)d_in[19]};

    const float* Wk = (const float*)d_in[20];
    const float* bk = (const float*)d_in[21];
    const float* Wq = (const float*)d_in[22];
    const float* bq = (const float*)d_in[23];
    const float* Wv = (const float*)d_in[24];
    const float* bv = (const float*)d_in[25];
    const float* Wa = (const float*)d_in[26];
    const float* ba = (const float*)d_in[27];
    const float* rel_att = (const float*)d_in[28];   // [L,R,NHEAD,DK,DK]
    const float* rel_msg = (const float*)d_in[29];
    const float* rel_pri = (const float*)d_in[30];   // [L,R,NHEAD]
    const float* skip    = (const float*)d_in[31];   // [L,T]
    const float* gamma   = (const float*)d_in[32];   // [L,H]
    const float* beta    = (const float*)d_in[33];   // [L,H]
    const int* edge_index = (const int*)d_in[35];    // [2,E]
    const int* edge_types = (const int*)d_in[36];    // [E]
    const int* src = edge_index;
    const int* dst = edge_index + EDGES;

    // ---- workspace carve-up ----
    char* ws = (char*)d_ws;
    size_t off = 0;
    auto carve = [&](size_t nfloats) -> float* {
        float* p = (float*)(ws + off);
        off += ((nfloats * sizeof(float)) + 255) & ~(size_t)255;
        return p;
    };
    float* tmp256 = carve((size_t)NPT * 256);        // encoder hidden
    float* hA     = carve((size_t)NTOT * HDIM);
    float* hB     = carve((size_t)NTOT * HDIM);
    float* kbuf   = carve((size_t)NTOT * HDIM);      // reused as `trans` later
    float* qbuf   = carve((size_t)NTOT * HDIM);
    float* vbuf   = carve((size_t)NTOT * HDIM);
    float* att    = carve((size_t)EDGES * NHEAD);    // att, then ea (in place)
    float* amax   = carve((size_t)NTOT * NHEAD);
    float* den    = carve((size_t)NTOT * NHEAD);
    float* res    = carve((size_t)NTOT * HDIM);
    (void)ws_size; (void)n_in; (void)in_sizes; (void)out_size;

    auto gemm = [&](const float* A, const float* W, const float* b, float* C,
                    int M, int K, int N, int act) {
        int blocks = ((M + 127) / 128) * (N / 64);
        gemm_bias_act<<<blocks, 256, 0, stream>>>(A, W, b, C, M, K, N, act);
    };

    // ---- per-type input encoders -> hA ----
    for (int t = 0; t < TYPES; ++t) {
        gemm(nf[t], ew1[t], eb1[t], tmp256, NPT, IN_DIMS[t], 256, /*relu*/1);
        gemm(tmp256, ew2[t], eb2[t], hA + (size_t)t * NPT * HDIM, NPT, 256, HDIM, 0);
    }

    // ---- HGT layers ----
    for (int l = 0; l < NLAYER; ++l) {
        const float* hin  = (l == 0) ? hA : hB;
        float*       hout = (l == 0) ? hB : (float*)d_out;
        const size_t WT   = (size_t)HDIM * HDIM;   // per (l,t) weight block

        for (int t = 0; t < TYPES; ++t) {
            const size_t wi = ((size_t)l * TYPES + t);
            const float* x  = hin + (size_t)t * NPT * HDIM;
            gemm(x, Wk + wi*WT, bk + wi*HDIM, kbuf + (size_t)t*NPT*HDIM, NPT, HDIM, HDIM, 0);
            gemm(x, Wq + wi*WT, bq + wi*HDIM, qbuf + (size_t)t*NPT*HDIM, NPT, HDIM, HDIM, 0);
            gemm(x, Wv + wi*WT, bv + wi*HDIM, vbuf + (size_t)t*NPT*HDIM, NPT, HDIM, HDIM, 0);
        }

        const long nNH = (long)NTOT * NHEAD;
        fill_f32<<<(int)((nNH + 255)/256), 256, 0, stream>>>(amax, -INFINITY, nNH);
        fill_f32<<<(int)((nNH + 255)/256), 256, 0, stream>>>(den, 0.f, nNH);
        fill_f32<<<(int)(((long)NTOT*HDIM + 255)/256), 256, 0, stream>>>(res, 0.f, (long)NTOT*HDIM);

        const size_t relOff = (size_t)l * RREL * NHEAD * DKH * DKH;
        dim3 egrid((EDGES + 255)/256, NHEAD);
        edge_att<<<egrid, 256, 0, stream>>>(qbuf, kbuf, src, dst, edge_types,
                                            rel_att + relOff,
                                            rel_pri + (size_t)l * RREL * NHEAD,
                                            att, amax, EDGES);
        edge_exp<<<(EDGES*NHEAD + 255)/256, 256, 0, stream>>>(att, dst, amax, den, EDGES*NHEAD);
        edge_scatter<<<egrid, 256, 0, stream>>>(vbuf, src, dst, edge_types,
                                                rel_msg + relOff, att, den, res, EDGES);

        gelu_inplace<<<(int)(((long)NTOT*HDIM + 255)/256), 256, 0, stream>>>(res, (long)NTOT*HDIM);

        float* trans = kbuf;  // k no longer needed this layer
        for (int t = 0; t < TYPES; ++t) {
            const size_t wi = ((size_t)l * TYPES + t);
            gemm(res + (size_t)t*NPT*HDIM, Wa + wi*WT, ba + wi*HDIM,
                 trans + (size_t)t*NPT*HDIM, NPT, HDIM, HDIM, 0);
        }

        node_update<<<(NTOT + 3)/4, 128, 0, stream>>>(trans, hin,
                                                      skip + (size_t)l*TYPES,
                                                      gamma + (size_t)l*HDIM,
                                                      beta + (size_t)l*HDIM,
                                                      hout, NTOT);
    }
}